// OccupancyGrid_36163624633057
// MI455X (gfx1250) — compile-verified
//
#include <hip/hip_runtime.h>
#include <hip/hip_bf16.h>
#include <stdint.h>

#define RES         256
#define THRESH      0.01f
#define THREADS     256
#define TILE_PTS    2048                       // points per staged tile
#define PTS_BYTES   (TILE_PTS * 12)            // 24576 B of point data per tile
#define DEN_BYTES   (TILE_PTS * 4)             // 8192 B of density data per tile
#define BUF_BYTES   (PTS_BYTES + DEN_BYTES)    // 32768 B per buffer (x2 for double buffer)
#define PTS_F4_TILE (PTS_BYTES / 16)           // 1536 float4 per tile
#define DEN_F4_TILE (DEN_BYTES / 16)           // 512 float4 per tile
#define PPT         (TILE_PTS / THREADS)       // 8 points per thread

// CDNA5 async global->LDS copy, 16B per lane, non-temporal load hint
// (input stream is single-use; keep the 192MB L2 for the 64MB grid which is RT).
__device__ __forceinline__ void async_ld_b128_nt(uint32_t lds_off, const void* gaddr) {
    asm volatile("global_load_async_to_lds_b128 %0, %1, off th:TH_LOAD_NT"
                 :: "v"(lds_off), "v"((unsigned long long)(uintptr_t)gaddr)
                 : "memory");
}
__device__ __forceinline__ void wait_async0() {
    asm volatile("s_wait_asynccnt 0" ::: "memory");
}

// Reproduce the reference's exact float sequence:
//   pts    = -1.0f + p * 2.0f            (rounded)
//   coordf = (pts - (-1.0f)) / 2 * 256   == (pts + 1.0f) * 128  (the *128 is exact)
//   coord  = clamp((int)coordf, 0, 255)
__device__ __forceinline__ int voxel_coord(float p) {
    float t = -1.0f + p * 2.0f;
    t = (t + 1.0f) * 128.0f;
    int i = (int)t;                       // truncation == .astype(int32) for non-negative
    i = i < 0 ? 0 : (i > RES - 1 ? RES - 1 : i);
    return i;
}

__global__ void zero_grid_kernel(float4* __restrict__ out, int n4) {
    int i = blockIdx.x * blockDim.x + threadIdx.x;
    if (i < n4) out[i] = make_float4(0.f, 0.f, 0.f, 0.f);
}

__global__ __launch_bounds__(THREADS)
void occ_scatter_kernel(const float4* __restrict__ ptsF4,
                        const float4* __restrict__ denF4,
                        float* __restrict__ out,
                        int N, int nTiles, int tilesPerBlock) {
    __shared__ float lds[2 * BUF_BYTES / 4];   // 64 KB double buffer
    const int tid = threadIdx.x;
    // generic shared address truncated to 32 bits == wave-relative LDS byte offset
    const uint32_t ldsBase = (uint32_t)(uintptr_t)(&lds[0]);

    const int ptsF4Total = (int)(((long long)N * 3 + 3) / 4);
    const int denF4Total = (N + 3) / 4;

    int t0 = blockIdx.x * tilesPerBlock;
    int t1 = t0 + tilesPerBlock;
    if (t1 > nTiles) t1 = nTiles;
    if (t0 >= t1) return;

    auto stage = [&](int tile, int buf) {
        const uint32_t b = ldsBase + (uint32_t)buf * BUF_BYTES;
        const int pBase = tile * PTS_F4_TILE;
        const int dBase = tile * DEN_F4_TILE;
#pragma unroll
        for (int k = 0; k < PTS_F4_TILE / THREADS; ++k) {        // 6 x b128
            int idx = pBase + k * THREADS + tid;
            if (idx < ptsF4Total)
                async_ld_b128_nt(b + (uint32_t)(k * THREADS + tid) * 16u, ptsF4 + idx);
        }
#pragma unroll
        for (int k = 0; k < DEN_F4_TILE / THREADS; ++k) {        // 2 x b128
            int idx = dBase + k * THREADS + tid;
            if (idx < denF4Total)
                async_ld_b128_nt(b + (uint32_t)PTS_BYTES + (uint32_t)(k * THREADS + tid) * 16u,
                                 denF4 + idx);
        }
    };

    stage(t0, 0);

    for (int t = t0; t < t1; ++t) {
        const int buf = (t - t0) & 1;

        wait_async0();        // my slice of tile t landed in LDS
        __syncthreads();      // everyone's slice landed

        if (t + 1 < t1) stage(t + 1, buf ^ 1);

        // L2 prefetch two tiles ahead (gfx1250 global_prefetch_b8)
        if (t + 2 < t1) {
            const char* pfp = (const char*)(ptsF4 + (long long)(t + 2) * PTS_F4_TILE);
            const char* pfd = (const char*)(denF4 + (long long)(t + 2) * DEN_F4_TILE);
            if (tid < PTS_BYTES / 128) __builtin_prefetch(pfp + tid * 128, 0, 0);
            else if (tid < PTS_BYTES / 128 + DEN_BYTES / 128)
                __builtin_prefetch(pfd + (tid - PTS_BYTES / 128) * 128, 0, 0);
        }

        // consume tile t from LDS
        const float* P = lds + buf * (BUF_BYTES / 4);
        const float* D = P + PTS_BYTES / 4;
        const int base = t * TILE_PTS;
#pragma unroll
        for (int k = 0; k < PPT; ++k) {
            const int p = k * THREADS + tid;
            if (base + p < N) {
                const float d = D[p];
                const float x = P[p * 3 + 0];
                const float y = P[p * 3 + 1];
                const float z = P[p * 3 + 2];
                if (d > THRESH) {
                    const int ix = voxel_coord(x);
                    const int iy = voxel_coord(y);
                    const int iz = voxel_coord(z);
                    // benign race: every writer stores the same 1.0f
                    out[(ix << 16) | (iy << 8) | iz] = 1.0f;
                }
            }
        }
        __syncthreads();      // all reads of this buffer done before it is restaged
    }
}

extern "C" void kernel_launch(void* const* d_in, const int* in_sizes, int n_in,
                              void* d_out, int out_size, void* d_ws, size_t ws_size,
                              hipStream_t stream) {
    const float4* ptsF4 = (const float4*)d_in[0];   // [N,3] fp32
    const float4* denF4 = (const float4*)d_in[1];   // [N]   fp32
    float* out = (float*)d_out;                     // 256^3 fp32
    const int N = in_sizes[1];

    // 1) zero the 64MB grid (it then lives dirty in the 192MB L2)
    const int n4 = out_size / 4;
    zero_grid_kernel<<<(n4 + THREADS - 1) / THREADS, THREADS, 0, stream>>>((float4*)d_out, n4);

    // 2) streaming scatter
    int nTiles = (N + TILE_PTS - 1) / TILE_PTS;
    int blocks = nTiles < 2048 ? nTiles : 2048;
    int tpb = (nTiles + blocks - 1) / blocks;
    blocks = (nTiles + tpb - 1) / tpb;
    occ_scatter_kernel<<<blocks, THREADS, 0, stream>>>(ptsF4, denF4, out, N, nTiles, tpb);
}